// DensityAwareChamferDistance_40913858462222
// MI455X (gfx1250) — compile-verified
//
#include <hip/hip_runtime.h>
#include <math.h>

typedef __attribute__((ext_vector_type(2))) float v2f;
typedef __attribute__((ext_vector_type(8))) float v8f;

#define ALPHA_   1000.0f
#define EPS_     1e-6f
#define BATCH    8
#define NPTS     4096      // N == M == 4096

// ---------------------------------------------------------------------------
// Kernel 1: per-point squared norms for both clouds + zero the output scalar.
// ---------------------------------------------------------------------------
__global__ void dacd_prep(const float* __restrict__ xyz1,
                          const float* __restrict__ xyz2,
                          float* __restrict__ sq1,
                          float* __restrict__ sq2,
                          float* __restrict__ out,
                          int total)
{
    int tid = blockIdx.x * blockDim.x + threadIdx.x;
    if (tid == 0) out[0] = 0.0f;
    if (tid < total) {
        const float* p = xyz1 + (size_t)tid * 3;
        sq1[tid] = p[0]*p[0] + p[1]*p[1] + p[2]*p[2];
        const float* q = xyz2 + (size_t)tid * 3;
        sq2[tid] = q[0]*q[0] + q[1]*q[1] + q[2]*q[2];
    }
}

// ---------------------------------------------------------------------------
// Kernel 2: brute-force argmin NN via V_WMMA_F32_16X16X4_F32.
//
// d_ij = |a|^2 + |b|^2 - 2 a.b folded into a single WMMA:
//   A row i = (-2x1, -2y1, -2z1, 1)   (loop-invariant, 2 VGPRs)
//   B col j = ( x2,   y2,   z2, |b|^2)
//   C_mn    = |a_m|^2                  (loop-invariant, 8 VGPRs)
//
// All 8 waves of a block share one batch; the B operands for all 256 column
// tiles are staged ONCE into LDS, pre-packed per (tile, lane) so the inner
// loop is exactly: 1x ds_load_b64 + 1x v_wmma + running argmin selects.
//
// ISA layouts (wave32):
//   A (16x4 f32): lanes 0-15 -> K=0,1 (x,y); lanes 16-31 -> K=2,3 (z, pad)
//   B (4x16 f32): mirrored (N = lane&15)
//   C/D (16x16 f32): VGPR r -> row r (lanes 0-15) / row r+8 (lanes 16-31)
// ---------------------------------------------------------------------------
__global__ __launch_bounds__(256)
void dacd_nn_argmin_wmma(const float* __restrict__ xyzA,
                         const float* __restrict__ sqA,
                         const float* __restrict__ xyzB,
                         const float* __restrict__ sqB,
                         int* __restrict__ idxOut)
{
    // 64 KB: packed B operands, lbuf[jt*32 + lane] = {K0,K1} (lanes 0-15) or {K2,K3} (lanes 16-31) pair for that lane
    __shared__ v2f lbuf[NPTS * 2];

    const int tid  = threadIdx.x;
    const int lane = tid & 31;
    const int wave = tid >> 5;
    const int gid  = blockIdx.x * 8 + wave;     // 2048 waves total
    const int b    = gid >> 8;                  // constant within a block
    const int t    = gid & 255;
    const int i0   = t << 4;
    const bool hi  = lane >= 16;
    const int  nl  = lane & 15;

    // --- stage packed B operands for this batch into LDS (once per block) ---
    {
        const float* pB = xyzB + (size_t)b * NPTS * 3;
        const float* pS = sqB  + (size_t)b * NPTS;
        for (int j = tid; j < NPTS; j += 256) {
            float x = pB[j*3+0], y = pB[j*3+1], z = pB[j*3+2];
            float s = pS[j];
            int jt = j >> 4, n = j & 15;
            v2f lo; lo.x = x; lo.y = y;         // lanes 0-15: K=0 (x), K=1 (y)
            v2f hv; hv.x = z; hv.y = s;         // lanes 16-31: K=2 (z), K=3 (|b|^2)
            lbuf[jt*32 + n]      = lo;
            lbuf[jt*32 + 16 + n] = hv;
        }
    }
    __syncthreads();

    // --- loop-invariant A operand -------------------------------------------
    const size_t baseA = ((size_t)b * NPTS + i0 + nl) * 3;
    v2f amat;
    amat.x = -2.0f * xyzA[baseA + (hi ? 2 : 0)];          // K=0 (x) / K=2 (z)
    amat.y = hi ? 1.0f : (-2.0f * xyzA[baseA + 1]);       // K=1 (y) / K=3 (pad=1)

    // --- loop-invariant C operand: C[r] = |a_row|^2, row = i0 + r + (hi?8:0) --
    v8f cmat;
    {
        const float* pc = sqA + (size_t)b * NPTS + i0 + (hi ? 8 : 0);
        #pragma unroll
        for (int r = 0; r < 8; ++r) cmat[r] = pc[r];
    }

    float minv[8];
    int   mini[8];
    #pragma unroll
    for (int r = 0; r < 8; ++r) { minv[r] = 3.4e38f; mini[r] = 0; }

    const v2f* lb = lbuf + lane;

    #pragma unroll 4
    for (int jt = 0; jt < NPTS / 16; ++jt) {
        v2f bmat = lb[jt * 32];                 // single ds_load_b64, bank-conflict-free

        // D = A x B + C  ->  full 16x16 tile of squared distances
        v8f d = __builtin_amdgcn_wmma_f32_16x16x4_f32(
            /*neg_a=*/false, amat, /*neg_b=*/false, bmat,
            /*c_mod=*/(short)0, cmat, /*reuse_a=*/false, /*reuse_b=*/false);

        const int j = jt * 16 + nl;
        #pragma unroll
        for (int r = 0; r < 8; ++r) {
            bool lt = d[r] < minv[r];
            minv[r] = lt ? d[r] : minv[r];
            mini[r] = lt ? j    : mini[r];
        }
    }

    // --- per-row argmin: butterfly across the 16 lanes sharing each row ------
    #pragma unroll
    for (int m = 8; m >= 1; m >>= 1) {
        #pragma unroll
        for (int r = 0; r < 8; ++r) {
            float ov = __shfl_xor(minv[r], m, 32);
            int   oi = __shfl_xor(mini[r], m, 32);
            bool take = (ov < minv[r]) || (ov == minv[r] && oi < mini[r]);
            minv[r] = take ? ov : minv[r];
            mini[r] = take ? oi : mini[r];
        }
    }

    if (nl == 0) {
        const int rowBase = i0 + (hi ? 8 : 0);
        #pragma unroll
        for (int r = 0; r < 8; ++r)
            idxOut[(size_t)b * NPTS + rowBase + r] = mini[r];
    }
}

// ---------------------------------------------------------------------------
// Kernel 3: per-batch epilogue.  LDS histogram of NN indices, recompute the
// gathered distance (as the reference does), density weight, exp, reduce,
// atomic-add the scaled partial into the scalar output.
// ---------------------------------------------------------------------------
__global__ __launch_bounds__(1024)
void dacd_finalize(const float* __restrict__ xyzA,
                   const float* __restrict__ xyzB,
                   const int* __restrict__ idx,
                   float* __restrict__ out,
                   float frac, float scale)
{
    __shared__ int   cnt[NPTS];     // 16 KB of 320 KB LDS
    __shared__ float red[32];

    const int b   = blockIdx.x;
    const int tid = threadIdx.x;

    for (int j = tid; j < NPTS; j += blockDim.x) cnt[j] = 0;
    __syncthreads();

    const int* ib = idx + (size_t)b * NPTS;
    for (int i = tid; i < NPTS; i += blockDim.x) atomicAdd(&cnt[ib[i]], 1);
    __syncthreads();

    const float* pa = xyzA + (size_t)b * NPTS * 3;
    const float* pb = xyzB + (size_t)b * NPTS * 3;

    float acc = 0.0f;
    for (int i = tid; i < NPTS; i += blockDim.x) {
        int j = ib[i];
        float dx = pa[i*3+0] - pb[j*3+0];
        float dy = pa[i*3+1] - pb[j*3+1];
        float dz = pa[i*3+2] - pb[j*3+2];
        float d  = dx*dx + dy*dy + dz*dz;
        float w  = frac / ((float)cnt[j] + EPS_);
        acc += 1.0f - __expf(-d * ALPHA_) * w;
    }

    // block reduction (wave32)
    #pragma unroll
    for (int off = 16; off >= 1; off >>= 1) acc += __shfl_down(acc, off, 32);
    if ((tid & 31) == 0) red[tid >> 5] = acc;
    __syncthreads();
    if (tid < 32) {
        float v = (tid < (int)(blockDim.x >> 5)) ? red[tid] : 0.0f;
        #pragma unroll
        for (int off = 16; off >= 1; off >>= 1) v += __shfl_down(v, off, 32);
        if (tid == 0) atomicAdd(out, v * scale);
    }
}

// ---------------------------------------------------------------------------
extern "C" void kernel_launch(void* const* d_in, const int* in_sizes, int n_in,
                              void* d_out, int out_size, void* d_ws, size_t ws_size,
                              hipStream_t stream)
{
    const float* xyz1 = (const float*)d_in[0];
    const float* xyz2 = (const float*)d_in[1];
    float* out = (float*)d_out;

    const int total = BATCH * NPTS;

    // workspace layout: sq1 | sq2 | idx1 | idx2   (4 * 32768 * 4B = 512 KB)
    float* sq1  = (float*)d_ws;
    float* sq2  = sq1 + total;
    int*   idx1 = (int*)(sq2 + total);
    int*   idx2 = idx1 + total;

    dacd_prep<<<(total + 255) / 256, 256, 0, stream>>>(xyz1, xyz2, sq1, sq2, out, total);

    // direction 1: for each xyz1 point, nearest in xyz2
    dacd_nn_argmin_wmma<<<256, 256, 0, stream>>>(xyz1, sq1, xyz2, sq2, idx1);
    // direction 2: swapped roles
    dacd_nn_argmin_wmma<<<256, 256, 0, stream>>>(xyz2, sq2, xyz1, sq1, idx2);

    // frac_21 = M/N = 1, frac_12 = N/M = 1 for N==M==4096
    const float scale = 1.0f / (2.0f * (float)BATCH * (float)NPTS);
    dacd_finalize<<<BATCH, 1024, 0, stream>>>(xyz1, xyz2, idx1, out, 1.0f, scale);
    dacd_finalize<<<BATCH, 1024, 0, stream>>>(xyz2, xyz1, idx2, out, 1.0f, scale);
}